// Net_47760036331718
// MI455X (gfx1250) — compile-verified
//
#include <hip/hip_runtime.h>
#include <hip/hip_bf16.h>

typedef __attribute__((ext_vector_type(2))) float v2f;
typedef __attribute__((ext_vector_type(4))) float v4f;
typedef __attribute__((ext_vector_type(8))) float v8f;

namespace {
constexpr int Bt = 256;   // batch
constexpr int Tt = 1024;  // time steps
constexpr int Dd = 64;    // input dim
constexpr int Hh = 100;   // hidden dim
constexpr int G4 = 400;   // 4*H gate columns
constexpr int BM = 16;    // batch rows per workgroup (WMMA M)
constexpr int NWAVES   = G4 / 16;      // 25 waves: one 16-col gate tile each
constexpr int NTHREADS = NWAVES * 32;  // 800 threads (wave32)
}

// Fused persistent LSTM scan. One workgroup = 16 batch rows, all 1024 steps.
// Per step, per wave:  acc(16x16) = bias + x_t(16x64)@W_ih^T + h(16x100)@W_hh^T
// via fp32 WMMA (16x16x4). Weights stay in VGPRs across the whole scan.
__global__ __launch_bounds__(NTHREADS)
void lstm_fused_scan(const float* __restrict__ x,     // [B,T,D]
                     const float* __restrict__ h0,    // [1,B,H]
                     const float* __restrict__ c0,    // [1,B,H]
                     const float* __restrict__ W_ih,  // [4H,D]
                     const float* __restrict__ W_hh,  // [4H,H]
                     const float* __restrict__ b_ih,  // [4H]
                     const float* __restrict__ b_hh,  // [4H]
                     const float* __restrict__ W_fc,  // [1,H]
                     const float* __restrict__ b_fc,  // [1]
                     float* __restrict__ out)         // [B,1]
{
  __shared__ float xbuf[2][BM][Dd];  // double-buffered x_t tile (8 KB)
  __shared__ float hbuf[BM][Hh];     // current hidden state (6.4 KB)
  __shared__ float cbuf[BM][Hh];     // current cell state   (6.4 KB)
  __shared__ float gbuf[BM][G4];     // gate pre-activations (25.6 KB)

  const int tid  = threadIdx.x;
  const int wave = tid >> 5;         // 0..24 -> gate-column tile
  const int lane = tid & 31;
  const int b0   = blockIdx.x * BM;  // this WG's batch rows

  // WMMA fp32 lane mapping (16x16x4):
  //  A: lane holds (m = lane&15, k = kbase + 2*(lane>>4) + {0,1})
  //  B: lane holds (n = nb + (lane&15), same k pair)  [mirrored layout]
  //  C/D: VGPR r -> row m = r + 8*(lane>>4), col n = nb + (lane&15)
  const int ncol  = wave * 16 + (lane & 15);   // global gate column 0..399
  const int khalf = (lane >> 4) * 2;           // 0 or 2
  const int am    = lane & 15;                 // A row
  const int mbase = (lane >> 4) * 8;           // C/D row base

  // ---- persistent B-matrix fragments in VGPRs (loaded once) ----
  v2f bih[16];                                  // W_ih^T, K=64 -> 16 chunks of 4
  #pragma unroll
  for (int kc = 0; kc < 16; ++kc)
    bih[kc] = *(const v2f*)(W_ih + ncol * Dd + kc * 4 + khalf);
  v2f bhh[25];                                  // W_hh^T, K=100 -> 25 chunks of 4
  #pragma unroll
  for (int kc = 0; kc < 25; ++kc)
    bhh[kc] = *(const v2f*)(W_hh + ncol * Hh + kc * 4 + khalf);
  const float bias = b_ih[ncol] + b_hh[ncol];   // per-column, folded into acc init

  // ---- init h, c from h0/c0; preload x tile for t=0 ----
  for (int e = tid; e < BM * Hh; e += NTHREADS) {
    int m = e / Hh, j = e % Hh;
    hbuf[m][j] = h0[(b0 + m) * Hh + j];
    cbuf[m][j] = c0[(b0 + m) * Hh + j];
  }
  int pm = 0, pd = 0;
  if (tid < 256) {                              // 16*64 floats = 256 lanes * float4
    pm = (tid * 4) >> 6;
    pd = (tid * 4) & 63;
    *(v4f*)&xbuf[0][pm][pd] =
        *(const v4f*)(x + ((size_t)(b0 + pm) * Tt) * Dd + pd);
  }
  __syncthreads();

  int cur = 0;
  for (int t = 0; t < Tt; ++t) {
    // -- prefetch x(t+1) into registers; committed to LDS after the barrier --
    v4f xnext = {0.f, 0.f, 0.f, 0.f};
    const bool pf = (tid < 256) && (t + 1 < Tt);
    if (pf)
      xnext = *(const v4f*)(x + ((size_t)(b0 + pm) * Tt + (t + 1)) * Dd + pd);

    // -- gates = bias + x_t @ W_ih^T + h @ W_hh^T  (two WMMA chains) --
    v8f acc0, acc1;
    #pragma unroll
    for (int r = 0; r < 8; ++r) { acc0[r] = bias; acc1[r] = 0.f; }

    #pragma unroll
    for (int kc = 0; kc < 16; kc += 2) {        // K = 64 (input term)
      v2f a0 = *(const v2f*)&xbuf[cur][am][kc * 4 + khalf];
      v2f a1 = *(const v2f*)&xbuf[cur][am][(kc + 1) * 4 + khalf];
      acc0 = __builtin_amdgcn_wmma_f32_16x16x4_f32(false, a0, false, bih[kc],
                                                   (short)0, acc0, false, false);
      acc1 = __builtin_amdgcn_wmma_f32_16x16x4_f32(false, a1, false, bih[kc + 1],
                                                   (short)0, acc1, false, false);
    }
    #pragma unroll
    for (int kc = 0; kc < 24; kc += 2) {        // K = 100 (recurrent term)
      v2f a0 = *(const v2f*)&hbuf[am][kc * 4 + khalf];
      v2f a1 = *(const v2f*)&hbuf[am][(kc + 1) * 4 + khalf];
      acc0 = __builtin_amdgcn_wmma_f32_16x16x4_f32(false, a0, false, bhh[kc],
                                                   (short)0, acc0, false, false);
      acc1 = __builtin_amdgcn_wmma_f32_16x16x4_f32(false, a1, false, bhh[kc + 1],
                                                   (short)0, acc1, false, false);
    }
    {
      v2f a = *(const v2f*)&hbuf[am][24 * 4 + khalf];
      acc0 = __builtin_amdgcn_wmma_f32_16x16x4_f32(false, a, false, bhh[24],
                                                   (short)0, acc0, false, false);
    }

    // -- scatter pre-activations to LDS (each (m,n) owned by one lane) --
    #pragma unroll
    for (int r = 0; r < 8; ++r)
      gbuf[mbase + r][ncol] = acc0[r] + acc1[r];
    __syncthreads();

    // -- elementwise gate math: 16*100 = 1600 elems over 800 threads --
    #pragma unroll
    for (int ee = 0; ee < 2; ++ee) {
      int e = tid + ee * NTHREADS;
      int m = e / Hh, j = e % Hh;
      float gi = gbuf[m][j];
      float gf = gbuf[m][j + Hh];
      float gg = gbuf[m][j + 2 * Hh];
      float go = gbuf[m][j + 3 * Hh];
      float si = 1.f / (1.f + expf(-gi));
      float sf = 1.f / (1.f + expf(-gf));
      float tg = tanhf(gg);
      float so = 1.f / (1.f + expf(-go));
      float cc = sf * cbuf[m][j] + si * tg;
      cbuf[m][j] = cc;
      hbuf[m][j] = so * tanhf(cc);
    }
    // commit prefetched x(t+1) into the other buffer
    if (pf) *(v4f*)&xbuf[cur ^ 1][pm][pd] = xnext;
    __syncthreads();
    cur ^= 1;
  }

  // -- FC head on final hidden state: out[b] = h_T[b] . W_fc + b_fc --
  if (tid < BM) {
    float s = b_fc[0];
    #pragma unroll 4
    for (int j = 0; j < Hh; ++j) s += hbuf[tid][j] * W_fc[j];
    out[b0 + tid] = s;
  }
}

extern "C" void kernel_launch(void* const* d_in, const int* in_sizes, int n_in,
                              void* d_out, int out_size, void* d_ws, size_t ws_size,
                              hipStream_t stream) {
  const float* x    = (const float*)d_in[0];
  const float* h0   = (const float*)d_in[1];
  const float* c0   = (const float*)d_in[2];
  const float* W_ih = (const float*)d_in[3];
  const float* W_hh = (const float*)d_in[4];
  const float* b_ih = (const float*)d_in[5];
  const float* b_hh = (const float*)d_in[6];
  const float* W_fc = (const float*)d_in[7];
  const float* b_fc = (const float*)d_in[8];
  float* out = (float*)d_out;

  dim3 grid(Bt / BM);      // 16 workgroups, one per 16 batch rows
  dim3 block(NTHREADS);    // 800 threads = 25 wave32 (one per gate tile)
  hipLaunchKernelGGL(lstm_fused_scan, grid, block, 0, stream,
                     x, h0, c0, W_ih, W_hh, b_ih, b_hh, W_fc, b_fc, out);
}